// GN_51891794870415
// MI455X (gfx1250) — compile-verified
//
#include <hip/hip_runtime.h>

// GraphSAGE mean-aggregate + dual GEMM, fused, fp32, CDNA5 WMMA.
//
// out = x @ W_self + mean_neigh(x) @ W_neigh + b
//   N = 100000 nodes, E = 1.25M edges (dst sorted), D = 64.
//
// One 64-thread block (2 wave32) per 16-node tile:
//   Phase 1: deterministic segmented mean over sorted dst -> h tile in LDS
//            (atomic-free: dst sorted => contiguous edge range per tile).
//   Phase 2: v_wmma_f32_16x16x4_f32 accumulation over K=64 for both GEMMs.

typedef __attribute__((ext_vector_type(2))) float v2f;
typedef __attribute__((ext_vector_type(8))) float v8f;

#define DIM 64
#define ROWS_PER_BLOCK 16
#define LDS_STRIDE 68   // 64 + 4 floats pad: lane bank = (4*m + k) % 64 -> conflict-free

__global__ __launch_bounds__(64, 8)
void sage_fused_wmma(const float* __restrict__ x,
                     const float* __restrict__ Wself,
                     const float* __restrict__ Wneigh,
                     const float* __restrict__ bias,
                     const int*   __restrict__ src,
                     const int*   __restrict__ dst,
                     float*       __restrict__ out,
                     int n_nodes, int n_edges)
{
    __shared__ float hlds[ROWS_PER_BLOCK * LDS_STRIDE];

    const int tid  = threadIdx.x;      // 0..63
    const int wave = tid >> 5;         // 0 or 1
    const int lane = tid & 31;         // 0..31
    const int row0 = blockIdx.x * ROWS_PER_BLOCK;

    // ---------------- Phase 1: segmented mean aggregation ----------------
    // Wave w owns rows [row0 + 8w, row0 + 8w + 8). dst is sorted, so the
    // edges for those rows are one contiguous range: two binary searches,
    // then an ordered linear scan (deterministic, atomic-free).
    {
        const int rbase = row0 + wave * 8;

        int lo = 0, hi = n_edges;
        while (lo < hi) { int mid = (lo + hi) >> 1; if (dst[mid] < rbase) lo = mid + 1; else hi = mid; }
        int p = lo;

        int lo2 = p, hi2 = n_edges;
        while (lo2 < hi2) { int mid = (lo2 + hi2) >> 1; if (dst[mid] < rbase + 8) lo2 = mid + 1; else hi2 = mid; }
        const int pend = lo2;

        for (int r = 0; r < 8; ++r) {
            const int v = rbase + r;
            float a0 = 0.f, a1 = 0.f, cnt = 0.f;
            // p, pend, v uniform across the wave -> convergent loop
            while (p < pend && dst[p] == v) {
                const int s = src[p];
                a0 += x[(long long)s * DIM + lane];        // 128B coalesced
                a1 += x[(long long)s * DIM + lane + 32];   // 128B coalesced
                cnt += 1.f;
                ++p;
            }
            const float rinv = 1.f / fmaxf(cnt, 1.f);      // mean; 0 if deg==0
            hlds[(v - row0) * LDS_STRIDE + lane]      = a0 * rinv;
            hlds[(v - row0) * LDS_STRIDE + lane + 32] = a1 * rinv;
        }
    }
    __syncthreads();

    // ---------------- Phase 2: 16x64 output tile via WMMA f32 ----------------
    // V_WMMA_F32_16X16X4_F32 operand mapping (wave32):
    //   A (16x4):  lanes 0-15 -> M=lane, K=k0+{0,1}; lanes 16-31 -> M=lane-16, K=k0+{2,3}
    //   B (4x16):  lanes 0-15 -> N=lane, K=k0+{0,1}; lanes 16-31 -> N=lane-16, K=k0+{2,3}
    //   C/D:       VGPR r: lanes 0-15 -> M=r, N=lane; lanes 16-31 -> M=r+8, N=lane-16
    const int  half = lane >> 4;
    const int  m    = lane & 15;
    const bool full = (row0 + ROWS_PER_BLOCK) <= n_nodes;   // block-uniform
    const int  arow = full ? (row0 + m)
                           : ((row0 + m < n_nodes) ? (row0 + m) : (n_nodes - 1));

    v8f acc[2];
    #pragma unroll
    for (int i = 0; i < 2; ++i) {
        const float bv = bias[(wave * 2 + i) * 16 + m];   // bias depends only on N
        acc[i] = (v8f){bv, bv, bv, bv, bv, bv, bv, bv};
    }

    #pragma unroll 4
    for (int k0 = 0; k0 < DIM; k0 += 4) {
        const int ka = k0 + 2 * half;

        v2f ax;                                           // A fragment of x (global)
        ax.x = x[(long long)arow * DIM + ka];
        ax.y = x[(long long)arow * DIM + ka + 1];

        v2f ah;                                           // A fragment of h (LDS)
        ah.x = hlds[m * LDS_STRIDE + ka];
        ah.y = hlds[m * LDS_STRIDE + ka + 1];

        #pragma unroll
        for (int i = 0; i < 2; ++i) {
            const int ncol = (wave * 2 + i) * 16 + m;
            v2f bs, bn;
            bs.x = Wself [(long long)ka * DIM + ncol];
            bs.y = Wself [(long long)(ka + 1) * DIM + ncol];
            bn.x = Wneigh[(long long)ka * DIM + ncol];
            bn.y = Wneigh[(long long)(ka + 1) * DIM + ncol];

            acc[i] = __builtin_amdgcn_wmma_f32_16x16x4_f32(
                         false, ax, false, bs, (short)0, acc[i], false, false);
            acc[i] = __builtin_amdgcn_wmma_f32_16x16x4_f32(
                         false, ah, false, bn, (short)0, acc[i], false, false);
        }
    }

    // ---------------- Store D ----------------
    // Hot path (every block when N % 16 == 0): no per-element guards, no exec
    // mask churn -- plain stores; lanes 0-15 / 16-31 each cover 64B contiguous
    // per r, and the 4 column tiles complete full 256B rows in L2.
    if (full) {
        #pragma unroll
        for (int i = 0; i < 2; ++i) {
            const int ncol = (wave * 2 + i) * 16 + m;
            #pragma unroll
            for (int r = 0; r < 8; ++r) {
                const int orow = row0 + r + 8 * half;
                out[(long long)orow * DIM + ncol] = acc[i][r];
            }
        }
    } else {
        #pragma unroll
        for (int i = 0; i < 2; ++i) {
            const int ncol = (wave * 2 + i) * 16 + m;
            #pragma unroll
            for (int r = 0; r < 8; ++r) {
                const int orow = row0 + r + 8 * half;
                if (orow < n_nodes)
                    out[(long long)orow * DIM + ncol] = acc[i][r];
            }
        }
    }
}

extern "C" void kernel_launch(void* const* d_in, const int* in_sizes, int n_in,
                              void* d_out, int out_size, void* d_ws, size_t ws_size,
                              hipStream_t stream) {
    const float* x   = (const float*)d_in[0];   // [N, 64]
    const float* Ws  = (const float*)d_in[1];   // [64, 64]
    const float* Wn  = (const float*)d_in[2];   // [64, 64]
    const float* b   = (const float*)d_in[3];   // [64]
    const int*   src = (const int*)  d_in[4];   // [E]
    const int*   dst = (const int*)  d_in[5];   // [E] sorted ascending

    float* out = (float*)d_out;                 // [N, 64]

    const int n_nodes = in_sizes[0] / DIM;
    const int n_edges = in_sizes[4];
    const int nblocks = (n_nodes + ROWS_PER_BLOCK - 1) / ROWS_PER_BLOCK;

    sage_fused_wmma<<<nblocks, 64, 0, stream>>>(x, Ws, Wn, b, src, dst, out,
                                                n_nodes, n_edges);
}